// CRF_68771016343786
// MI455X (gfx1250) — compile-verified
//
#include <hip/hip_runtime.h>
#include <hip/hip_bf16.h>

// Problem constants (from the reference)
#define BB 256
#define SS 1024
#define TT 128
#define TILE_B 16
#define LSTR 132   // padded LDS row stride (floats): conflict-free transposed reads

typedef float v8f __attribute__((ext_vector_type(8)));
typedef float v2f __attribute__((ext_vector_type(2)));

// Order-preserving float <-> uint encoding so LDS u32 atomicMax == float max.
__device__ __forceinline__ unsigned fenc(float f) {
    unsigned u = __float_as_uint(f);
    return (u & 0x80000000u) ? ~u : (u | 0x80000000u);
}
__device__ __forceinline__ float fdec(unsigned k) {
    unsigned u = (k & 0x80000000u) ? (k & 0x7FFFFFFFu) : ~k;
    return __uint_as_float(u);
}

// ---------------------------------------------------------------------------
// Kernel 1: gold path score per batch (cheap, O(B*S) scattered gathers)
// ---------------------------------------------------------------------------
__global__ __launch_bounds__(256) void crf_score_kernel(
    const float* __restrict__ em, const int* __restrict__ tags,
    const unsigned char* __restrict__ mask, const float* __restrict__ trans,
    const float* __restrict__ startT, const float* __restrict__ endT,
    float* __restrict__ scoreOut)
{
    int b = blockIdx.x * blockDim.x + threadIdx.x;
    if (b >= BB) return;
    const int* tg = &tags[(long)b * SS];
    const unsigned char* mk = &mask[(long)b * SS];
    int t0 = tg[0];
    float s = startT[t0] + em[((long)b * SS) * TT + t0];
    int cnt = mk[0] ? 1 : 0;
    for (int t = 1; t < SS; ++t) {
        int tp = tg[t - 1], tc = tg[t];
        float mf = mk[t] ? 1.0f : 0.0f;
        s += mf * (trans[tp * TT + tc] + em[((long)b * SS + t) * TT + tc]);
        cnt += mk[t] ? 1 : 0;
    }
    s += endT[tg[cnt - 1]];
    scoreOut[b] = s;
}

// ---------------------------------------------------------------------------
// Kernel 2: forward algorithm.  One WG per 16-batch tile, 8 waves = 8 state
// column tiles.  exp(trans) B-fragments live in VGPRs for the whole scan.
// Per step: exp into LDS -> 32x v_wmma_f32_16x16x4_f32 (4 independent
// accumulator chains) -> log+emit epilogue with ds-atomic row-max for the
// next step's rescaling.  Only 2 barriers per step.
// ---------------------------------------------------------------------------
__global__ __launch_bounds__(256) void crf_forward_kernel(
    const float* __restrict__ em, const unsigned char* __restrict__ mask,
    const float* __restrict__ trans, const float* __restrict__ startT,
    float* __restrict__ alphaOut)
{
    __shared__ float    alphaS[TILE_B * LSTR];  // log-domain alpha
    __shared__ float    expA[TILE_B * LSTR];    // rescaled exp(alpha - rowmax)
    __shared__ unsigned rmaxE[2][TILE_B];       // encoded row maxes (dbl-buffered)
    __shared__ float    cmaxS[TILE_B];          // decoded per-row max
    __shared__ int      maskS[TILE_B];          // mask bits for this step

    const int tid  = threadIdx.x;
    const int lane = tid & 31;
    const int wave = tid >> 5;
    const int b0   = blockIdx.x * TILE_B;
    const int n0   = wave * 16;      // this wave's state-column tile
    const int nl   = lane & 15;
    const int hi   = lane >> 4;      // 0: K offsets 0/1, 1: K offsets 2/3

    // ---- B operand: exp(transitions) column tile, resident in VGPRs ----
    // 16x16x4 B layout (4xK by 16xN): VGPR0 holds K = {0 | 2}, VGPR1 = {1 | 3}
    v2f bfrag[32];
#pragma unroll
    for (int c = 0; c < 32; ++c) {
        int k0 = 4 * c + 2 * hi;
        bfrag[c].x = __expf(trans[(long)k0 * TT + n0 + nl]);
        bfrag[c].y = __expf(trans[(long)(k0 + 1) * TT + n0 + nl]);
    }

    if (tid < 2 * TILE_B) rmaxE[tid >> 4][tid & 15] = 0u;   // encoded -inf
    __syncthreads();

    // ---- alpha0 = start_transitions + emissions[:, 0, :] ----
    {
        int row = tid >> 4, cbase = (tid & 15) * 8;
        float mloc = -3.4e38f;
#pragma unroll
        for (int i = 0; i < 8; ++i) {
            int c = cbase + i;
            float v = startT[c] + em[((long)(b0 + row) * SS) * TT + c];
            alphaS[row * LSTR + c] = v;
            mloc = fmaxf(mloc, v);
        }
        atomicMax(&rmaxE[1][row], fenc(mloc));   // step t=1 reads buffer (t&1)=1
    }
    __syncthreads();

    for (int t = 1; t < SS; ++t) {
        const int cur = t & 1, nxt = (t + 1) & 1;

        // Phase A: read row max, reset next-step max buffer, exp into expA
        {
            int row = tid >> 4, cg = tid & 15;
            float m = fdec(rmaxE[cur][row]);
            if (cg == 0) { cmaxS[row] = m; rmaxE[nxt][row] = 0u; }
#pragma unroll
            for (int i = 0; i < 8; ++i) {
                int c = cg * 8 + i;
                expA[row * LSTR + c] = __expf(alphaS[row * LSTR + c] - m);
            }
            if (tid < TILE_B) maskS[tid] = mask[(long)(b0 + tid) * SS + t];
            if (t + 1 < SS)   // warm L2/L0 for next step's emissions
                __builtin_prefetch(&em[((long)(b0 + row) * SS + (t + 1)) * TT + cg * 8], 0, 1);
        }
        __syncthreads();

        // Phase B: C(16x16) = expA(16x128) * expT(128x16), 32 WMMAs across
        // 4 independent accumulator chains (shorter RAW critical path).
        v8f A0 = {}, A1 = {}, A2 = {}, A3 = {};
#pragma unroll
        for (int c = 0; c < 32; c += 4) {
            int k0 = 4 * c + 2 * hi;
            v2f a0, a1, a2, a3;
            a0.x = expA[nl * LSTR + k0];      a0.y = expA[nl * LSTR + k0 + 1];
            a1.x = expA[nl * LSTR + k0 + 4];  a1.y = expA[nl * LSTR + k0 + 5];
            a2.x = expA[nl * LSTR + k0 + 8];  a2.y = expA[nl * LSTR + k0 + 9];
            a3.x = expA[nl * LSTR + k0 + 12]; a3.y = expA[nl * LSTR + k0 + 13];
            A0 = __builtin_amdgcn_wmma_f32_16x16x4_f32(false, a0, false, bfrag[c + 0], (short)0, A0, false, false);
            A1 = __builtin_amdgcn_wmma_f32_16x16x4_f32(false, a1, false, bfrag[c + 1], (short)0, A1, false, false);
            A2 = __builtin_amdgcn_wmma_f32_16x16x4_f32(false, a2, false, bfrag[c + 2], (short)0, A2, false, false);
            A3 = __builtin_amdgcn_wmma_f32_16x16x4_f32(false, a3, false, bfrag[c + 3], (short)0, A3, false, false);
        }
        v8f C = (A0 + A1) + (A2 + A3);

        // Epilogue: newAlpha = log(C) + rowmax + emit; masked write-back;
        // feed next step's row max via LDS u32 atomic max (encoded float).
        {
            const int mbase = hi * 8;  // C/D layout: VGPR v -> M = v (+8 hi half)
#pragma unroll
            for (int v = 0; v < 8; ++v) {
                int M = v + mbase;
                float s = fmaxf(C[v], 1e-37f);
                float val = __logf(s) + cmaxS[M] +
                            em[((long)(b0 + M) * SS + t) * TT + n0 + nl];
                float sv;
                if (maskS[M]) { alphaS[M * LSTR + n0 + nl] = val; sv = val; }
                else          { sv = alphaS[M * LSTR + n0 + nl]; }
                atomicMax(&rmaxE[nxt][M], fenc(sv));
            }
        }
        __syncthreads();
    }

    // Write final alpha to workspace
    {
        int row = tid >> 4, cbase = (tid & 15) * 8;
#pragma unroll
        for (int i = 0; i < 8; ++i)
            alphaOut[(long)(b0 + row) * TT + cbase + i] = alphaS[row * LSTR + cbase + i];
    }
}

// ---------------------------------------------------------------------------
// Kernel 3: log_z = LSE(alpha + end_transitions); out = mean(log_z - score)
// ---------------------------------------------------------------------------
__global__ __launch_bounds__(256) void crf_final_kernel(
    const float* __restrict__ alphaG, const float* __restrict__ endT,
    const float* __restrict__ scoreG, float* __restrict__ out)
{
    __shared__ float red[256];
    int b = threadIdx.x;
    const float* a = &alphaG[(long)b * TT];
    float m = a[0] + endT[0];
    for (int j = 1; j < TT; ++j) m = fmaxf(m, a[j] + endT[j]);
    float ssum = 0.0f;
    for (int j = 0; j < TT; ++j) ssum += __expf(a[j] + endT[j] - m);
    float logz = __logf(ssum) + m;
    red[b] = logz - scoreG[b];
    __syncthreads();
    for (int off = 128; off > 0; off >>= 1) {
        if (b < off) red[b] += red[b + off];
        __syncthreads();
    }
    if (b == 0) out[0] = red[0] * (1.0f / BB);
}

// ---------------------------------------------------------------------------
extern "C" void kernel_launch(void* const* d_in, const int* in_sizes, int n_in,
                              void* d_out, int out_size, void* d_ws, size_t ws_size,
                              hipStream_t stream) {
    (void)in_sizes; (void)n_in; (void)out_size; (void)ws_size;
    const float*         em     = (const float*)d_in[0];         // (B,S,T) f32
    const int*           tags   = (const int*)d_in[1];           // (B,S) i32
    const unsigned char* mask   = (const unsigned char*)d_in[2]; // (B,S) bool
    const float*         trans  = (const float*)d_in[3];         // (T,T)
    const float*         startT = (const float*)d_in[4];         // (T,)
    const float*         endT   = (const float*)d_in[5];         // (T,)
    float*               out    = (float*)d_out;                 // scalar

    float* scores = (float*)d_ws;                          // 256 floats
    float* alphaG = (float*)((char*)d_ws + 4096);          // B*T floats (128 KB)

    crf_score_kernel<<<1, 256, 0, stream>>>(em, tags, mask, trans, startT, endT, scores);
    crf_forward_kernel<<<BB / TILE_B, 256, 0, stream>>>(em, mask, trans, startT, alphaG);
    crf_final_kernel<<<1, 256, 0, stream>>>(alphaG, endT, scores, out);
}